// GCNModel_85555748536563
// MI455X (gfx1250) — compile-verified
//
#include <hip/hip_runtime.h>
#include <hip/hip_bf16.h>
#include <math.h>

#define HD 128          // hidden width
#define IND 5           // input features
#define GG 64           // graphs
#define BN_EPS 1e-5f

typedef float v2f __attribute__((ext_vector_type(2)));
typedef float v8f __attribute__((ext_vector_type(8)));

__device__ __forceinline__ void atomAddF(float* p, float v) {
    (void)__hip_atomic_fetch_add(p, v, __ATOMIC_RELAXED, __HIP_MEMORY_SCOPE_AGENT);
}

__device__ __forceinline__ float siluf(float x) {
    return x / (1.0f + __expf(-x));
}

// ---------------------------------------------------------------- utilities
__global__ void zero_kernel(float* p, int n) {
    int i = blockIdx.x * blockDim.x + threadIdx.x;
    if (i < n) p[i] = 0.0f;
}

// deg[col[e]] += ew[e]
__global__ void deg_kernel(const int* __restrict__ col, const float* __restrict__ ew,
                           float* __restrict__ deg, int E) {
    int e = blockIdx.x * blockDim.x + threadIdx.x;
    if (e < E) atomAddF(&deg[col[e]], ew[e]);
}

// dinv[i] = rsqrt(deg[i] + 1)   (self-loop weight 1 -> deg always > 0), in place
__global__ void dinv_kernel(float* __restrict__ deg, int n) {
    int i = blockIdx.x * blockDim.x + threadIdx.x;
    if (i < n) deg[i] = rsqrtf(deg[i] + 1.0f);
}

// norm[e] = dinv[row]*ew*dinv[col]
__global__ void norm_kernel(const int* __restrict__ row, const int* __restrict__ col,
                            const float* __restrict__ ew, const float* __restrict__ dinv,
                            float* __restrict__ nrm, int E) {
    int e = blockIdx.x * blockDim.x + threadIdx.x;
    if (e < E) nrm[e] = dinv[row[e]] * ew[e] * dinv[col[e]];
}

// ------------------------------------------------- layer-0 transform (K=5)
__global__ void transform0_kernel(const float* __restrict__ x, const float* __restrict__ w0,
                                  float* __restrict__ m, int n) {
    int idx = blockIdx.x * blockDim.x + threadIdx.x;
    if (idx >= n * HD) return;
    int i = idx >> 7, j = idx & (HD - 1);
    const float* xi = x + (size_t)i * IND;
    float acc = 0.0f;
#pragma unroll
    for (int k = 0; k < IND; ++k) acc += xi[k] * w0[k * HD + j];
    m[idx] = acc;
}

// ------------------------------------------------- WMMA GEMM: M = A(n x 128) * W(128 x 128)
// 256 threads = 8 waves; wave w owns rows [blk*128 + w*16, +16).
// W staged in LDS in K-pair-interleaved layout so each B fragment is one
// aligned ds_load_b64 directly into the WMMA operand pair:
//   lw[(p*HD + c)*2 + 0] = W[2p  ][c]
//   lw[(p*HD + c)*2 + 1] = W[2p+1][c]     (p = k/2, c = column)
__global__ __launch_bounds__(256) void gemm_h128_kernel(const float* __restrict__ A,
                                                        const float* __restrict__ W,
                                                        float* __restrict__ Mout, int n) {
    __shared__ float lw[HD * HD];
    for (int t = threadIdx.x; t < (HD / 2) * HD; t += 256) {
        int p = t >> 7;        // K-pair index 0..63
        int c = t & (HD - 1);  // column
        float2 v;
        v.x = W[(2 * p) * HD + c];
        v.y = W[(2 * p + 1) * HD + c];
        *(float2*)(lw + 2 * t) = v;   // 2*t == (p*HD + c)*2
    }
    __syncthreads();

    const int wave = threadIdx.x >> 5;
    const int lane = threadIdx.x & 31;
    const int rowBase = blockIdx.x * 128 + wave * 16;
    if (rowBase >= n) return;              // wave-uniform exit (N is a multiple of 16)

    const int m     = rowBase + (lane & 15);
    const int khalf = (lane >> 4) << 1;    // 0 or 2 (A/B fragment K split)
    const int bcol  = lane & 15;

    v8f acc[8] = {};
    for (int k0 = 0; k0 < HD; k0 += 4) {
        const int p = (k0 + khalf) >> 1;   // K-pair index for this lane half
        v2f a;
        const float2 av = *(const float2*)(A + (size_t)m * HD + k0 + khalf);
        a.x = av.x; a.y = av.y;
#pragma unroll
        for (int t = 0; t < 8; ++t) {
            v2f b;
            const float2 bv = *(const float2*)(lw + (size_t)(p * HD + t * 16 + bcol) * 2);
            b.x = bv.x; b.y = bv.y;
            acc[t] = __builtin_amdgcn_wmma_f32_16x16x4_f32(
                false, a, false, b, (short)0, acc[t], false, false);
        }
    }
    // D layout: VGPR j -> row rowBase + j (+8 for lanes 16..31), col = tile*16 + (lane&15)
    const int rOff = (lane >= 16) ? 8 : 0;
#pragma unroll
    for (int t = 0; t < 8; ++t) {
#pragma unroll
        for (int j = 0; j < 8; ++j) {
            int r = rowBase + j + rOff;
            Mout[(size_t)r * HD + t * 16 + (lane & 15)] = acc[t][j];
        }
    }
}

// ------------------------------------------------- agg init: h = bias + dinv^2 * m (self loop)
__global__ void agginit_kernel(const float* __restrict__ m, const float* __restrict__ dinv,
                               const float* __restrict__ bias, float* __restrict__ h, int n) {
    int idx = blockIdx.x * blockDim.x + threadIdx.x;
    if (idx >= n * 32) return;
    int i = idx >> 5, j = (idx & 31) * 4;
    float ds = dinv[i] * dinv[i];
    float4 v = *(const float4*)(m + (size_t)i * HD + j);
    float4 o;
    o.x = bias[j + 0] + ds * v.x;
    o.y = bias[j + 1] + ds * v.y;
    o.z = bias[j + 2] + ds * v.z;
    o.w = bias[j + 3] + ds * v.w;
    *(float4*)(h + (size_t)i * HD + j) = o;
}

// ------------------------------------------------- edge scatter: h[col] += norm * m[row]
// one wave32 per edge, one float4 per lane, 4 f32 atomics
__global__ __launch_bounds__(256) void scatter_kernel(const float* __restrict__ m,
                                                      const int* __restrict__ row,
                                                      const int* __restrict__ col,
                                                      const float* __restrict__ nrm,
                                                      float* __restrict__ h, int E) {
    int wid  = (blockIdx.x * blockDim.x + threadIdx.x) >> 5;
    int lane = threadIdx.x & 31;
    int nw   = (gridDim.x * blockDim.x) >> 5;
    for (int e = wid; e < E; e += nw) {
        int r = row[e], c = col[e];
        float w = nrm[e];
        const float4 v = *(const float4*)(m + (size_t)r * HD + lane * 4);
        float* dst = h + (size_t)c * HD + lane * 4;
        atomAddF(dst + 0, w * v.x);
        atomAddF(dst + 1, w * v.y);
        atomAddF(dst + 2, w * v.z);
        atomAddF(dst + 3, w * v.w);
    }
}

// ------------------------------------------------- BN column stats (sum, sumsq)
__global__ __launch_bounds__(256) void bnstats_kernel(const float* __restrict__ h,
                                                      float* __restrict__ colsum,
                                                      float* __restrict__ colsq, int n) {
    __shared__ float ls[256], lq[256];
    int j    = threadIdx.x & (HD - 1);
    int half = threadIdx.x >> 7;
    float s = 0.0f, q = 0.0f;
    for (int i = blockIdx.x * 2 + half; i < n; i += gridDim.x * 2) {
        float v = h[(size_t)i * HD + j];
        s += v; q += v * v;
    }
    ls[threadIdx.x] = s; lq[threadIdx.x] = q;
    __syncthreads();
    if (threadIdx.x < HD) {
        s = ls[threadIdx.x] + ls[threadIdx.x + HD];
        q = lq[threadIdx.x] + lq[threadIdx.x + HD];
        atomAddF(&colsum[j], s);
        atomAddF(&colsq[j], q);
    }
}

__global__ void bnfinalize_kernel(const float* __restrict__ colsum, const float* __restrict__ colsq,
                                  const float* __restrict__ gamma, const float* __restrict__ beta,
                                  float* __restrict__ scale, float* __restrict__ shift, int n) {
    int j = threadIdx.x;
    if (j < HD) {
        float invn = 1.0f / (float)n;
        float mu  = colsum[j] * invn;
        float var = colsq[j] * invn - mu * mu;
        float is  = rsqrtf(var + BN_EPS);
        float sc  = gamma[j] * is;
        scale[j] = sc;
        shift[j] = beta[j] - mu * sc;
    }
}

// fused BN-apply + SiLU (in place)
__global__ void bnapply_kernel(float* __restrict__ h, const float* __restrict__ scale,
                               const float* __restrict__ shift, int n) {
    int idx = blockIdx.x * blockDim.x + threadIdx.x;
    if (idx >= n * 32) return;
    int i = idx >> 5, j = (idx & 31) * 4;
    float4 v = *(const float4*)(h + (size_t)i * HD + j);
    float4 o;
    o.x = siluf(v.x * scale[j + 0] + shift[j + 0]);
    o.y = siluf(v.y * scale[j + 1] + shift[j + 1]);
    o.z = siluf(v.z * scale[j + 2] + shift[j + 2]);
    o.w = siluf(v.w * scale[j + 3] + shift[j + 3]);
    *(float4*)(h + (size_t)i * HD + j) = o;
}

// ------------------------------------------------- mean pool (atomic accumulate)
__global__ __launch_bounds__(256) void pool_kernel(const float* __restrict__ h,
                                                   const int* __restrict__ batch,
                                                   float* __restrict__ pooled,
                                                   float* __restrict__ counts, int n) {
    int wid  = (blockIdx.x * blockDim.x + threadIdx.x) >> 5;
    int lane = threadIdx.x & 31;
    int nw   = (gridDim.x * blockDim.x) >> 5;
    for (int i = wid; i < n; i += nw) {
        int g = batch[i];
        const float4 v = *(const float4*)(h + (size_t)i * HD + lane * 4);
        float* dst = pooled + (size_t)g * HD + lane * 4;
        atomAddF(dst + 0, v.x);
        atomAddF(dst + 1, v.y);
        atomAddF(dst + 2, v.z);
        atomAddF(dst + 3, v.w);
        if (lane == 0) atomAddF(&counts[g], 1.0f);
    }
}

// ------------------------------------------------- MLP head: silu(fc1) -> sigmoid(fc2)
__global__ void head_kernel(const float* __restrict__ pooled, const float* __restrict__ counts,
                            const float* __restrict__ fc1w, const float* __restrict__ fc1b,
                            const float* __restrict__ fc2w, const float* __restrict__ fc2b,
                            float* __restrict__ out) {
    __shared__ float p[HD];
    __shared__ float f1[64];
    int g = blockIdx.x, t = threadIdx.x;
    float cnt = counts[g];
    cnt = cnt > 1.0f ? cnt : 1.0f;
    if (t < HD) p[t] = pooled[(size_t)g * HD + t] / cnt;
    __syncthreads();
    if (t < 64) {
        float s = fc1b[t];
        for (int k = 0; k < HD; ++k) s += p[k] * fc1w[k * 64 + t];
        f1[t] = siluf(s);
    }
    __syncthreads();
    if (t == 0) {
        float s = fc2b[0];
        for (int k = 0; k < 64; ++k) s += f1[k] * fc2w[k];
        out[g] = 1.0f / (1.0f + __expf(-s));
    }
}

// ===========================================================================
extern "C" void kernel_launch(void* const* d_in, const int* in_sizes, int n_in,
                              void* d_out, int out_size, void* d_ws, size_t ws_size,
                              hipStream_t stream) {
    const float* x      = (const float*)d_in[0];
    const int*   eidx   = (const int*)d_in[1];      // [2, E]: row then col
    const float* eattr  = (const float*)d_in[2];
    const int*   batch  = (const int*)d_in[3];
    const float* w0     = (const float*)d_in[5];    // (5,128)
    const float* b0     = (const float*)d_in[6];
    const float* convW  = (const float*)d_in[7];    // (3,128,128)
    const float* convB  = (const float*)d_in[8];    // (3,128)
    const float* gamma  = (const float*)d_in[9];    // (4,128)
    const float* beta   = (const float*)d_in[10];
    const float* fc1w   = (const float*)d_in[11];
    const float* fc1b   = (const float*)d_in[12];
    const float* fc2w   = (const float*)d_in[13];
    const float* fc2b   = (const float*)d_in[14];
    float* out = (float*)d_out;

    const int E = in_sizes[2];
    const int N = in_sizes[3];

    // workspace layout
    float* ws = (float*)d_ws;
    size_t off = 0;
    float* mbuf   = ws + off; off += (size_t)N * HD;   // transformed features
    float* hbuf   = ws + off; off += (size_t)N * HD;   // activations / agg target
    float* dinv   = ws + off; off += (size_t)N;        // deg -> dinv in place
    float* nrm    = ws + off; off += (size_t)E;        // edge norms
    float* colsum = ws + off; off += HD;
    float* colsq  = ws + off; off += HD;
    float* scale  = ws + off; off += HD;
    float* shift  = ws + off; off += HD;
    float* pooled = ws + off; off += (size_t)GG * HD;  // pooled + counts contiguous
    float* counts = ws + off; off += GG;
    (void)ws_size; (void)n_in; (void)out_size;

    const int* row = eidx;
    const int* col = eidx + E;

    const int T = 256;
    const int gElemNH = (N * HD + T - 1) / T;
    const int gElem32 = (N * 32 + T - 1) / T;
    const int gEdge1  = (E + T - 1) / T;
    const int gNode1  = (N + T - 1) / T;
    const int gGemm   = (N + 127) / 128;

    // ---- gcn_norm
    zero_kernel<<<gNode1, T, 0, stream>>>(dinv, N);
    deg_kernel<<<gEdge1, T, 0, stream>>>(col, eattr, dinv, E);
    dinv_kernel<<<gNode1, T, 0, stream>>>(dinv, N);
    norm_kernel<<<gEdge1, T, 0, stream>>>(row, col, eattr, dinv, nrm, E);

    for (int layer = 0; layer < 4; ++layer) {
        // transform
        if (layer == 0) {
            transform0_kernel<<<gElemNH, T, 0, stream>>>(x, w0, mbuf, N);
        } else {
            gemm_h128_kernel<<<gGemm, T, 0, stream>>>(hbuf, convW + (size_t)(layer - 1) * HD * HD,
                                                      mbuf, N);
        }
        const float* bias = (layer == 0) ? b0 : (convB + (size_t)(layer - 1) * HD);
        // aggregate: self loops + bias, then edge scatter
        agginit_kernel<<<gElem32, T, 0, stream>>>(mbuf, dinv, bias, hbuf, N);
        scatter_kernel<<<2048, T, 0, stream>>>(mbuf, row, col, nrm, hbuf, E);
        // batchnorm (training) + silu
        zero_kernel<<<1, T, 0, stream>>>(colsum, 2 * HD);   // colsum & colsq contiguous
        bnstats_kernel<<<256, T, 0, stream>>>(hbuf, colsum, colsq, N);
        bnfinalize_kernel<<<1, HD, 0, stream>>>(colsum, colsq, gamma + (size_t)layer * HD,
                                                beta + (size_t)layer * HD, scale, shift, N);
        bnapply_kernel<<<gElem32, T, 0, stream>>>(hbuf, scale, shift, N);
    }

    // ---- pool + head
    zero_kernel<<<(GG * HD + GG + T - 1) / T, T, 0, stream>>>(pooled, GG * HD + GG);
    pool_kernel<<<1024, T, 0, stream>>>(hbuf, batch, pooled, counts, N);
    head_kernel<<<GG, HD, 0, stream>>>(pooled, counts, fc1w, fc1b, fc2w, fc2b, out);
}